// GaussianFalloffKernelVorticity_84421877170797
// MI455X (gfx1250) — compile-verified
//
#include <hip/hip_runtime.h>

typedef float v2f __attribute__((ext_vector_type(2)));
typedef float v8f __attribute__((ext_vector_type(8)));

#define NVORT 128   // N
#define HWPTS (256 * 256)

// One wave32 computes vorticity for 16 consecutive points, summing all 128
// vortices via 8x V_WMMA_F32_16X16X4_F32 (sq-dist as a rank-4 bilinear form).
__global__ __launch_bounds__(256)
void vorticity_wmma_kernel(const float* __restrict__ vf,   // [B,128,4] (y,x,tau,sig)
                           const float* __restrict__ pts,  // [B,H,W,2]
                           float* __restrict__ out)        // [B,H,W,1]
{
    const int lane = threadIdx.x & 31;
    const int nlo  = lane & 15;   // column / point-row selector within half
    const int half = lane >> 4;   // 0: lanes 0-15, 1: lanes 16-31

    const int wave = (blockIdx.x << 3) + (threadIdx.x >> 5);
    const int base = wave << 4;        // first of 16 flat point indices
    const int b    = base >> 16;       // / (256*256), tiles never cross batches

    // ---- A matrix (16x4 f32): row m = (py, px, |p|^2, 1) ----
    // Lane layout: lanes 0-15 hold K={0,1}, lanes 16-31 hold K={2,3}, m = lane&15.
    const float2 p = reinterpret_cast<const float2*>(pts)[base + nlo];
    v2f A;
    A.x = half ? (p.x * p.x + p.y * p.y) : p.x;   // K0 / K2
    A.y = half ? 1.0f                    : p.y;   // K1 / K3

    // ---- B matrices (4x16 f32) + per-column params, all 8 N-tiles ----
    // B col n = (-2ly, -2lx, 1, |loc|^2); VGPR0 = rows K0/K2, VGPR1 = K1/K3.
    v2f   Bm[8];
    float sc[8], cf[8];
    #pragma unroll
    for (int t = 0; t < 8; ++t) {
        const float4 v = reinterpret_cast<const float4*>(vf)[b * NVORT + t * 16 + nlo];
        const float ly = v.x, lx = v.y, tau = v.z, sig = v.w;
        const float sig2 = sig * sig;
        const float inv  = __builtin_amdgcn_rcpf(sig2);
        sc[t] = inv * 1.44269504088896340736f;        // log2(e)/sig^2
        cf[t] = tau * inv * 0.31830988618379067154f;  // tau/(pi*sig^2)
        Bm[t].x = half ? 1.0f                : -2.0f * ly;  // K0 / K2
        Bm[t].y = half ? (ly * ly + lx * lx) : -2.0f * lx;  // K1 / K3
    }

    // ---- main loop: WMMA -> per-element exp2 -> accumulate ----
    float acc[8] = {0.f, 0.f, 0.f, 0.f, 0.f, 0.f, 0.f, 0.f};
    const v8f zero = {};
    #pragma unroll
    for (int t = 0; t < 8; ++t) {
        // D[m][n] = py*(-2ly) + px*(-2lx) + |p|^2 + |loc|^2 = sq_dist
        v8f d = __builtin_amdgcn_wmma_f32_16x16x4_f32(
            false, A, false, Bm[t], (short)0, zero, false, false);
        #pragma unroll
        for (int r = 0; r < 8; ++r) {
            acc[r] += cf[t] * __builtin_amdgcn_exp2f(-d[r] * sc[t]);
        }
    }

    // ---- reduce the 16 vortex columns (lanes) within each half ----
    #pragma unroll
    for (int r = 0; r < 8; ++r) {
        float v = acc[r];
        v += __shfl_xor(v, 1, 32);
        v += __shfl_xor(v, 2, 32);
        v += __shfl_xor(v, 4, 32);
        v += __shfl_xor(v, 8, 32);
        acc[r] = v;   // all lanes in half hold totals for M = r + 8*half
    }

    // ---- lanes 0 and 16 store rows 0-7 / 8-15 as two B128 stores ----
    if (nlo == 0) {
        float* o = out + base + half * 8;   // 32-byte aligned
        reinterpret_cast<float4*>(o)[0] = make_float4(acc[0], acc[1], acc[2], acc[3]);
        reinterpret_cast<float4*>(o)[1] = make_float4(acc[4], acc[5], acc[6], acc[7]);
    }
}

extern "C" void kernel_launch(void* const* d_in, const int* in_sizes, int n_in,
                              void* d_out, int out_size, void* d_ws, size_t ws_size,
                              hipStream_t stream) {
    (void)in_sizes; (void)n_in; (void)out_size; (void)d_ws; (void)ws_size;
    const float* vf  = (const float*)d_in[0];   // [8,128,4]
    const float* pts = (const float*)d_in[1];   // [8,256,256,2]
    float* out = (float*)d_out;                 // [8,256,256,1]

    const int total_points = 8 * HWPTS;         // 524288
    const int waves  = total_points / 16;       // 32768 (16 pts per wave)
    const int blocks = waves / 8;               // 4096 (8 waves per block)
    vorticity_wmma_kernel<<<blocks, 256, 0, stream>>>(vf, pts, out);
}